// SwitchTransformersSparseMLPOffloading_29858612642047
// MI455X (gfx1250) — compile-verified
//
#include <hip/hip_runtime.h>
#include <stdint.h>

#define B_    4
#define S_    1024
#define DM    512
#define DFF   2048
#define NE    8
#define TOPK  2
#define NTOK  (B_ * S_)        // 4096
#define NPAIR (NTOK * TOPK)    // 8192
#define BM    16               // gathered rows per block
#define FCH   128              // D_FF chunk width

typedef __attribute__((ext_vector_type(2))) float v2f;
typedef __attribute__((ext_vector_type(8))) float v8f;

__device__ __forceinline__ v8f wmma_f32(v2f a, v2f b, v8f c) {
    // V_WMMA_F32_16X16X4_F32 : D = A(16x4) * B(4x16) + C(16x16), fp32
    return __builtin_amdgcn_wmma_f32_16x16x4_f32(
        /*neg_a=*/false, a, /*neg_b=*/false, b,
        /*c_mod=*/(short)0, c, /*reuse_a=*/false, /*reuse_b=*/false);
}

// ---------------- routing / setup kernels ----------------

__global__ void k_zero(float* __restrict__ out, int n,
                       int* __restrict__ counts, int* __restrict__ cursors) {
    int i = blockIdx.x * blockDim.x + threadIdx.x;
    if (i < n) out[i] = 0.0f;
    if (blockIdx.x == 0 && threadIdx.x < NE) {
        counts[threadIdx.x]  = 0;
        cursors[threadIdx.x] = 0;
    }
}

__global__ void k_count(const int* __restrict__ eidx, int* __restrict__ counts) {
    int p = blockIdx.x * blockDim.x + threadIdx.x;
    if (p < NPAIR) atomicAdd(&counts[eidx[p]], 1);
}

__global__ void k_scan(const int* __restrict__ counts, int* __restrict__ offs) {
    if (threadIdx.x == 0 && blockIdx.x == 0) {
        int s = 0;
        for (int e = 0; e < NE; ++e) { offs[e] = s; s += counts[e]; }
        offs[NE] = s;
    }
}

__global__ void k_fill(const int* __restrict__ eidx, const float* __restrict__ prob,
                       const int* __restrict__ offs, int* __restrict__ cursors,
                       int* __restrict__ tokens, float* __restrict__ wgts) {
    int p = blockIdx.x * blockDim.x + threadIdx.x;
    if (p >= NPAIR) return;
    int e   = eidx[p];
    int pos = offs[e] + atomicAdd(&cursors[e], 1);
    int k   = p & (TOPK - 1);
    tokens[pos] = p / TOPK;
    wgts[pos]   = ((k == 0) ? 0.5f : 0.25f) * prob[p];
}

// ---------------- fused per-expert gather-GEMM ----------------
// grid = (NPAIR/BM, NE), block = 256 threads (8 wave32)

__global__ __launch_bounds__(256, 1)
void k_moe_gemm(const float* __restrict__ hidden, const float* __restrict__ wi,
                const float* __restrict__ wo, const int* __restrict__ offs,
                const int* __restrict__ tokens, const float* __restrict__ wgts,
                float* __restrict__ acc) {
    __shared__ float Xs[BM][DM + 4];   // 16 x 516 : pad -> bank-conflict-free, f4-aligned
    __shared__ float As[BM][FCH + 4];  // 16 x 132 : relu(X @ wi) chunk
    __shared__ int   tokS[BM];
    __shared__ float wS[BM];

    const int e         = blockIdx.y;
    const int base      = offs[e];
    const int nrows_all = offs[e + 1] - base;
    const int rbase     = blockIdx.x * BM;
    if (rbase >= nrows_all) return;
    const int nrows = min(BM, nrows_all - rbase);

    const int tid = threadIdx.x;
    if (tid < BM) {
        int r = rbase + tid;
        if (r < nrows_all) { tokS[tid] = tokens[base + r]; wS[tid] = wgts[base + r]; }
        else               { tokS[tid] = 0;                wS[tid] = 0.0f; }
    }
    __syncthreads();

    // ---- async gather of X tile straight into LDS (ASYNCcnt path, no VGPR bounce)
    // 16 rows x 512 f32; 16 threads/row; each thread moves 8 x B128
    {
        const int row = tid >> 4;
        const int l16 = tid & 15;
        uint64_t g   = (uint64_t)(uintptr_t)(hidden + (size_t)tokS[row] * DM + l16 * 32);
        uint32_t lds = (uint32_t)(uintptr_t)&Xs[row][l16 * 32];  // LDS aperture: low 32 bits = LDS offset
        #pragma unroll
        for (int i = 0; i < 8; ++i) {
            asm volatile("global_load_async_to_lds_b128 %0, %1, off"
                         :: "v"(lds + i * 16), "v"(g + (uint64_t)(i * 16))
                         : "memory");
        }
        asm volatile("s_wait_asynccnt 0x0" ::: "memory");
    }
    __syncthreads();

    const int wave  = tid >> 5;   // 0..7
    const int lane  = tid & 31;
    const int lrow  = lane & 15;  // M (A) or N (B/C) index
    const int khalf = lane >> 4;  // which K pair

    const float* wiE = wi + (size_t)e * DM * DFF;
    const float* woE = wo + (size_t)e * DFF * DM;

    const v8f vzero = {0.f, 0.f, 0.f, 0.f, 0.f, 0.f, 0.f, 0.f};
    v8f acc2[4];                  // wave's 16 x 64 slice of the 16 x 512 output
    #pragma unroll
    for (int j = 0; j < 4; ++j) acc2[j] = vzero;

    const int nbase = wave * 64;  // GEMM2 N-slice per wave

    for (int fc = 0; fc < DFF; fc += FCH) {
        if (fc + FCH < DFF)
            __builtin_prefetch(woE + (size_t)(fc + FCH) * DM + nbase, 0, 1);

        // ---- GEMM1: this wave computes A1 tile [16 x 16] at N = fc + wave*16
        v8f c1 = vzero;
        const int bcol = fc + wave * 16 + lrow;
        #pragma unroll 8
        for (int k0 = 0; k0 < DM; k0 += 4) {
            const int ka = k0 + 2 * khalf;
            v2f a = *(const v2f*)&Xs[lrow][ka];
            v2f b;
            b.x = wiE[(size_t)ka * DFF + bcol];
            b.y = wiE[(size_t)(ka + 1) * DFF + bcol];
            c1 = wmma_f32(a, b, c1);
        }
        __syncthreads();  // previous chunk's GEMM2 reads of As are done
        #pragma unroll
        for (int r = 0; r < 8; ++r) {
            float v = c1[r];
            As[khalf * 8 + r][wave * 16 + lrow] = (v > 0.0f) ? v : 0.0f;  // relu
        }
        __syncthreads();

        // ---- GEMM2: C[16 x 64 slice] += As[16 x 128] @ wo[fc:fc+128, nbase:nbase+64]
        #pragma unroll 4
        for (int k0 = 0; k0 < FCH; k0 += 4) {
            const int ka = k0 + 2 * khalf;
            v2f a = *(const v2f*)&As[lrow][ka];
            const float* wrow0 = woE + (size_t)(fc + ka) * DM;
            const float* wrow1 = wrow0 + DM;
            #pragma unroll
            for (int j = 0; j < 4; ++j) {
                const int n = nbase + j * 16 + lrow;
                v2f b;
                b.x = wrow0[n];
                b.y = wrow1[n];
                acc2[j] = wmma_f32(a, b, acc2[j]);
            }
        }
    }

    // ---- scaled scatter-add (<=2 commutative f32 adds per output element -> deterministic)
    #pragma unroll
    for (int j = 0; j < 4; ++j) {
        const int n = nbase + j * 16 + lrow;
        #pragma unroll
        for (int r = 0; r < 8; ++r) {
            const int m = khalf * 8 + r;
            if (m < nrows) {
                atomicAdd(&acc[(size_t)tokS[m] * DM + n], acc2[j][r] * wS[m]);
            }
        }
    }
}

__global__ void k_finalize(const float* __restrict__ hidden, float* __restrict__ out, int n) {
    int i = blockIdx.x * blockDim.x + threadIdx.x;
    if (i < n) {
        float a = out[i];
        out[i] = (a != 0.0f) ? a : hidden[i];
    }
}

// ---------------- launch ----------------

extern "C" void kernel_launch(void* const* d_in, const int* in_sizes, int n_in,
                              void* d_out, int out_size, void* d_ws, size_t ws_size,
                              hipStream_t stream) {
    const float* hidden = (const float*)d_in[0];  // [4,1024,512]
    const float* prob   = (const float*)d_in[1];  // [4,1024,2]
    const float* wi     = (const float*)d_in[2];  // [8,512,2048]
    const float* wo     = (const float*)d_in[3];  // [8,2048,512]
    const int*   eidx   = (const int*)d_in[4];    // [4,1024,2]
    float* out = (float*)d_out;                   // [4,1024,512]

    char* ws      = (char*)d_ws;
    int*  counts  = (int*)ws;                 // 8
    int*  cursors = counts + NE;              // 8
    int*  offs    = cursors + NE;             // 9
    int*  tokens  = (int*)(ws + 128);         // 8192
    float* wgts   = (float*)(ws + 128 + NPAIR * sizeof(int));

    const int nOut = NTOK * DM;

    k_zero<<<(nOut + 255) / 256, 256, 0, stream>>>(out, nOut, counts, cursors);
    k_count<<<(NPAIR + 255) / 256, 256, 0, stream>>>(eidx, counts);
    k_scan<<<1, 32, 0, stream>>>(counts, offs);
    k_fill<<<(NPAIR + 255) / 256, 256, 0, stream>>>(eidx, prob, offs, cursors, tokens, wgts);
    k_moe_gemm<<<dim3(NPAIR / BM, NE), 256, 0, stream>>>(hidden, wi, wo, offs, tokens, wgts, out);
    k_finalize<<<(nOut + 255) / 256, 256, 0, stream>>>(hidden, out, nOut);
}